// Sparse_attention_764504178978
// MI455X (gfx1250) — compile-verified
//
#include <hip/hip_runtime.h>
#include <cstdint>
#include <cfloat>

// Problem dims are fixed by the reference: attn_raw is (MEM, BATCH) fp32.
// Both are powers of two -> all index math is shifts/masks, no division.
#define MEM_DIM    4096
#define BATCH_DIM  16384
#define BATCH_SH   14            // log2(BATCH_DIM)
#define TOP_K      8
#define SPLIT      8             // row-chunks for pass-1 parallelism (4096 waves)
#define ROWS_CHUNK (MEM_DIM / SPLIT)   // 512
#define EPS_F      1e-7f
#define BLOCK      256

typedef float v4f __attribute__((ext_vector_type(4)));

// Branch-light sorted insert: v[0] >= v[1] >= ... >= v[7].
// Only enters the compare-exchange chain when x beats the current 8th value.
__device__ __forceinline__ void topk_insert(float (&v)[TOP_K], float x) {
    if (x > v[TOP_K - 1]) {
        v[TOP_K - 1] = x;
#pragma unroll
        for (int i = TOP_K - 1; i > 0; --i) {
            float a = v[i - 1], b = v[i];
            v[i - 1] = fmaxf(a, b);
            v[i]     = fminf(a, b);
        }
    }
}

// ---------------------------------------------------------------------------
// Pass 1: per (column, row-chunk) partial top-8. Coalesced: consecutive
// threads handle consecutive columns of the same row. 8x row unroll for MLP
// (8 outstanding b32 loads/lane), speculative prefetch 16 rows ahead.
// ---------------------------------------------------------------------------
__global__ void pass1_partial_topk(const float* __restrict__ A,
                                   float* __restrict__ part) {
    int tid   = blockIdx.x * blockDim.x + threadIdx.x;   // 0 .. BATCH*SPLIT-1
    int col   = tid & (BATCH_DIM - 1);
    int chunk = tid >> BATCH_SH;

    const size_t stride = (size_t)BATCH_DIM;
    const float* p = A + (size_t)(chunk * ROWS_CHUNK) * stride + col;

    float v[TOP_K];
#pragma unroll
    for (int i = 0; i < TOP_K; ++i) v[i] = -FLT_MAX;

#pragma unroll 1
    for (int r = 0; r < ROWS_CHUNK; r += 8) {
        float x0 = p[0];
        float x1 = p[stride];
        float x2 = p[2 * stride];
        float x3 = p[3 * stride];
        float x4 = p[4 * stride];
        float x5 = p[5 * stride];
        float x6 = p[6 * stride];
        float x7 = p[7 * stride];
        __builtin_prefetch(p + 16 * stride, 0, 1);
        p += 8 * stride;
        topk_insert(v, x0);
        topk_insert(v, x1);
        topk_insert(v, x2);
        topk_insert(v, x3);
        topk_insert(v, x4);
        topk_insert(v, x5);
        topk_insert(v, x6);
        topk_insert(v, x7);
    }

    // part layout: [(chunk*TOP_K + k) * BATCH + col] -> coalesced R/W
#pragma unroll
    for (int k = 0; k < TOP_K; ++k)
        part[(size_t)(chunk * TOP_K + k) * stride + col] = v[k];
}

// ---------------------------------------------------------------------------
// Pass 2: merge SPLIT*TOP_K candidates per column -> delta and 1/(sum+eps).
// The clipped-sum only involves the top-8 values themselves (everything at or
// below v8 clips to zero since delta = v8 + eps), so no extra data pass.
// ---------------------------------------------------------------------------
__global__ void pass2_merge(const float* __restrict__ part,
                            float* __restrict__ delta,
                            float* __restrict__ inv) {
    int col = blockIdx.x * blockDim.x + threadIdx.x;

    float v[TOP_K];
#pragma unroll
    for (int i = 0; i < TOP_K; ++i) v[i] = -FLT_MAX;

#pragma unroll 8
    for (int c = 0; c < SPLIT * TOP_K; ++c)
        topk_insert(v, part[(size_t)c * BATCH_DIM + col]);

    float d = v[TOP_K - 1] + EPS_F;
    float s = 0.f;
#pragma unroll
    for (int i = 0; i < TOP_K; ++i) s += fmaxf(v[i] - d, 0.f);

    delta[col] = d;
    inv[col]   = 1.f / (s + EPS_F);
}

// ---------------------------------------------------------------------------
// Pass 3: streaming normalize, traversed in REVERSE address order so the
// first reads hit the lines pass 1 touched last (still resident in the
// 192 MB L2). One float4 per thread; the attn tile is staged through LDS via
// the CDNA5 async path (global_load_async_to_lds_b128, non-temporal, +
// s_wait_asynccnt), read back with ds_load_b128, result leaves via a
// non-temporal b128 store (output is never re-read).
// ---------------------------------------------------------------------------
__global__ void pass3_normalize(const float* __restrict__ A,
                                const float* __restrict__ delta,
                                const float* __restrict__ inv,
                                float* __restrict__ out) {
    __shared__ __align__(16) float lds_buf[BLOCK * 4];

    const size_t N4 = (size_t)MEM_DIM * BATCH_DIM / 4;      // 16M float4s
    size_t tid = (size_t)blockIdx.x * blockDim.x + threadIdx.x;
    size_t i4  = (N4 - 1) - tid;                            // reverse traversal
    int    c4  = (int)(i4 & ((BATCH_DIM >> 2) - 1));        // column group

    // Issue async b128 copy global -> this lane's 16B LDS slot (last use: NT).
    // Low 32 bits of a flat LDS-aperture pointer ARE the LDS byte address.
    uint64_t gaddr = (uint64_t)(uintptr_t)(A + i4 * 4);
    uint32_t laddr = (uint32_t)(uintptr_t)(&lds_buf[threadIdx.x * 4]);
    asm volatile("global_load_async_to_lds_b128 %0, %1, off th:TH_LOAD_NT"
                 :: "v"(laddr), "v"(gaddr) : "memory");

    // Overlap: fetch per-column delta / inv while the async copy is in flight.
    v4f d  = *(const v4f*)(delta + (size_t)c4 * 4);
    v4f iv = *(const v4f*)(inv   + (size_t)c4 * 4);

    asm volatile("s_wait_asynccnt 0x0" ::: "memory");
    v4f x = *(const v4f*)(&lds_buf[threadIdx.x * 4]);

    v4f r;
    r.x = fmaxf(x.x - d.x, 0.f) * iv.x;
    r.y = fmaxf(x.y - d.y, 0.f) * iv.y;
    r.z = fmaxf(x.z - d.z, 0.f) * iv.z;
    r.w = fmaxf(x.w - d.w, 0.f) * iv.w;

    __builtin_nontemporal_store(r, (v4f*)(out + i4 * 4));
}

// ---------------------------------------------------------------------------
extern "C" void kernel_launch(void* const* d_in, const int* in_sizes, int n_in,
                              void* d_out, int out_size, void* d_ws, size_t ws_size,
                              hipStream_t stream) {
    const float* A   = (const float*)d_in[0];
    float*       out = (float*)d_out;
    (void)in_sizes; (void)n_in; (void)out_size; (void)ws_size;

    // Workspace layout (floats): [SPLIT*TOP_K*BATCH partials][BATCH delta][BATCH inv]
    float* wsf   = (float*)d_ws;
    float* part  = wsf;
    float* delta = wsf + (size_t)SPLIT * TOP_K * BATCH_DIM;
    float* inv   = delta + BATCH_DIM;

    // All sizes are exact multiples of BLOCK (power-of-two dims).
    pass1_partial_topk<<<(BATCH_DIM * SPLIT) / BLOCK, BLOCK, 0, stream>>>(A, part);
    pass2_merge<<<BATCH_DIM / BLOCK, BLOCK, 0, stream>>>(part, delta, inv);

    const size_t N4 = (size_t)MEM_DIM * BATCH_DIM / 4;
    pass3_normalize<<<(unsigned)(N4 / BLOCK), BLOCK, 0, stream>>>(A, delta, inv, out);
}